// ContrastiveLoss_76733885710824
// MI455X (gfx1250) — compile-verified
//
#include <hip/hip_runtime.h>
#include <hip/hip_bf16.h>

// ---------------------------------------------------------------------------
// ContrastiveLoss fused kernel for MI455X (gfx1250, wave32, WMMA f32 path).
//
//   S = (F/||F||) @ (L/||L||)^T        [8192 x 4096], D = 128
//   loss = mean_i( cnt_i>0 ? -(sum_j lab*S - cnt_i*log(sum_j exp(S)))/cnt_i : 0 )
//   (TAU cancels analytically: log(exp(S)/t) - log(sum exp(S)/t) = S - LSE)
// ---------------------------------------------------------------------------

typedef __attribute__((ext_vector_type(2))) float v2f;
typedef __attribute__((ext_vector_type(8))) float v8f;

#define B_ROWS 8192
#define L_COLS 4096
#define DDIM   128
#define SPLIT  4
#define COLS_PER_SPLIT (L_COLS / SPLIT)   // 1024
#define BM     64                          // rows per workgroup

// ---------------------------------------------------------------------------
// Kernel 1: inverse L2 norms for both embedding tables. One wave per row.
// ---------------------------------------------------------------------------
__global__ __launch_bounds__(256) void norm_kernel(
    const float* __restrict__ feat, const float* __restrict__ lenc,
    float* __restrict__ invNf, float* __restrict__ invNl)
{
    const int w    = blockIdx.x * 8 + (threadIdx.x >> 5);   // global wave id
    const int lane = threadIdx.x & 31;

    const float* src;
    float*       dst;
    int          row;
    if (w < B_ROWS) { src = feat; dst = invNf; row = w; }
    else            { src = lenc; dst = invNl; row = w - B_ROWS; }

    const float4 v = *(const float4*)(src + (size_t)row * DDIM + lane * 4);
    float s = v.x * v.x + v.y * v.y + v.z * v.z + v.w * v.w;
    #pragma unroll
    for (int m = 16; m >= 1; m >>= 1) s += __shfl_xor(s, m, 32);
    if (lane == 0) dst[row] = 1.0f / fmaxf(sqrtf(s), 1e-8f);
}

// ---------------------------------------------------------------------------
// Kernel 2: fused GEMM (f32 WMMA 16x16x4) + streaming softmax-style row sums.
// Block = 256 threads = 8 waves: 4 row-tiles (64 rows) x 2 col-tiles (32 cols
// per iteration). grid = (B/64, SPLIT). Each split covers 1024 columns.
// Per row we accumulate: sum(exp(S)), sum(lab*S), sum(lab).
// ---------------------------------------------------------------------------
__global__ __launch_bounds__(256) void gemm_loss_kernel(
    const float* __restrict__ feat, const float* __restrict__ lenc,
    const int*   __restrict__ labels,
    const float* __restrict__ invNf, const float* __restrict__ invNl,
    float* __restrict__ pExp, float* __restrict__ pLS, float* __restrict__ pCnt)
{
    const int tid     = threadIdx.x;
    const int wave    = tid >> 5;        // 0..7
    const int lane    = tid & 31;
    const int rowTile = wave & 3;        // 4 row tiles of 16
    const int colTile = wave >> 2;       // 2 col tiles of 16

    const int blockRow = blockIdx.x * BM;
    const int colBase  = blockIdx.y * COLS_PER_SPLIT;

    // A fragment addressing (ISA 7.12.2, 32-bit A 16x4):
    //   lanes 0-15: M=lane,   K = {0,1};  lanes 16-31: M=lane-16, K = {2,3}
    const int m     = lane & 15;
    const int kHalf = (lane & 16) ? 2 : 0;
    const int r0    = blockRow + rowTile * 16;
    const int rowA  = r0 + m;

    // Hold this wave's 16x128 A slice in registers: 32 K-chunks of 4.
    v2f aFrag[32];
    #pragma unroll
    for (int kk = 0; kk < 32; ++kk)
        aFrag[kk] = *(const v2f*)(feat + (size_t)rowA * DDIM + kk * 4 + kHalf);

    // C/D layout: lane<16 -> rows r0+0..7 (vgpr v = row offset), lane>=16 -> r0+8..15
    const int rbase = r0 + ((lane & 16) ? 8 : 0);
    float invF[8];
    #pragma unroll
    for (int v = 0; v < 8; ++v) invF[v] = invNf[rbase + v];

    float accExp[8], accLS[8], accCnt[8];
    #pragma unroll
    for (int v = 0; v < 8; ++v) { accExp[v] = 0.f; accLS[v] = 0.f; accCnt[v] = 0.f; }

    for (int cb = 0; cb < COLS_PER_SPLIT; cb += 32) {
        const int c0   = colBase + cb + colTile * 16;
        const int colB = c0 + m;                    // this lane's B column

        v8f acc = {};
        #pragma unroll
        for (int kk = 0; kk < 32; ++kk) {
            // B (4x16) fragment: lane<16 holds K={0,1}, lane>=16 K={2,3} of col colB.
            // B[k][n] = label_enc[n*D + k]  ->  contiguous float2 per lane.
            v2f b = *(const v2f*)(lenc + (size_t)colB * DDIM + kk * 4 + kHalf);
            acc = __builtin_amdgcn_wmma_f32_16x16x4_f32(
                /*neg_a=*/false, aFrag[kk], /*neg_b=*/false, b,
                /*c_mod=*/(short)0, acc, /*reuse_a=*/false, /*reuse_b=*/false);
        }

        const float invL = invNl[colB];
        #pragma unroll
        for (int v = 0; v < 8; ++v) {
            const int row = rbase + v;
            const int lidx = row * L_COLS + colB;
            const float lf = (float)labels[lidx];
            // prefetch next column tile of the 128MB labels stream
            __builtin_prefetch(&labels[lidx + 32], 0, 1);
            const float s = acc[v] * invF[v] * invL;   // cosine similarity
            accExp[v] += __expf(s);
            accLS[v]  += lf * s;
            accCnt[v] += lf;
        }
    }

    // Reduce across the 16 lanes of each half (columns handled by this wave).
    #pragma unroll
    for (int v = 0; v < 8; ++v) {
        #pragma unroll
        for (int mm = 8; mm >= 1; mm >>= 1) {
            accExp[v] += __shfl_xor(accExp[v], mm, 32);
            accLS[v]  += __shfl_xor(accLS[v],  mm, 32);
            accCnt[v] += __shfl_xor(accCnt[v], mm, 32);
        }
    }

    // Cross-wave combine via LDS; each (colTile,row) slot written exactly once.
    __shared__ float sExp[2][BM], sLS[2][BM], sCnt[2][BM];
    if ((lane & 15) == 0) {
        const int rl = rowTile * 16 + ((lane & 16) ? 8 : 0);
        #pragma unroll
        for (int v = 0; v < 8; ++v) {
            sExp[colTile][rl + v] = accExp[v];
            sLS [colTile][rl + v] = accLS[v];
            sCnt[colTile][rl + v] = accCnt[v];
        }
    }
    __syncthreads();

    if (tid < BM) {
        const size_t o = (size_t)blockIdx.y * B_ROWS + blockRow + tid;
        pExp[o] = sExp[0][tid] + sExp[1][tid];
        pLS [o] = sLS [0][tid] + sLS [1][tid];
        pCnt[o] = sCnt[0][tid] + sCnt[1][tid];
    }
}

// ---------------------------------------------------------------------------
// Kernel 3: deterministic finalize. Single block: combine the SPLIT column
// partials per row, compute per-row loss, tree-reduce, write mean.
// ---------------------------------------------------------------------------
__global__ __launch_bounds__(256) void finalize_kernel(
    const float* __restrict__ pExp, const float* __restrict__ pLS,
    const float* __restrict__ pCnt, float* __restrict__ out)
{
    float local = 0.f;
    for (int r = threadIdx.x; r < B_ROWS; r += 256) {
        float se = 0.f, sl = 0.f, c = 0.f;
        #pragma unroll
        for (int s = 0; s < SPLIT; ++s) {
            se += pExp[(size_t)s * B_ROWS + r];
            sl += pLS [(size_t)s * B_ROWS + r];
            c  += pCnt[(size_t)s * B_ROWS + r];
        }
        if (c > 0.f) {
            const float lse = __logf(se);
            local += -(sl - c * lse) / fmaxf(c, 1.f);
        }
    }
    __shared__ float red[256];
    red[threadIdx.x] = local;
    __syncthreads();
    #pragma unroll
    for (int s = 128; s > 0; s >>= 1) {
        if (threadIdx.x < s) red[threadIdx.x] += red[threadIdx.x + s];
        __syncthreads();
    }
    if (threadIdx.x == 0) out[0] = red[0] / (float)B_ROWS;
}

// ---------------------------------------------------------------------------
extern "C" void kernel_launch(void* const* d_in, const int* in_sizes, int n_in,
                              void* d_out, int out_size, void* d_ws, size_t ws_size,
                              hipStream_t stream) {
    const float* feat   = (const float*)d_in[0];   // [8192,128] f32
    const float* lenc   = (const float*)d_in[1];   // [4096,128] f32
    const int*   labels = (const int*)d_in[2];     // [8192,4096] i32

    float* ws    = (float*)d_ws;
    float* invNf = ws;                             // 8192
    float* invNl = ws + B_ROWS;                    // 4096
    float* pExp  = ws + B_ROWS + L_COLS;           // SPLIT * 8192
    float* pLS   = pExp + (size_t)SPLIT * B_ROWS;
    float* pCnt  = pLS  + (size_t)SPLIT * B_ROWS;

    norm_kernel<<<(B_ROWS + L_COLS) / 8 / 32 * 32 / 8, 256, 0, stream>>>(
        feat, lenc, invNf, invNl);  // 192 blocks x 8 waves = 1536 rows

    dim3 grid(B_ROWS / BM, SPLIT);  // 128 x 4
    gemm_loss_kernel<<<grid, 256, 0, stream>>>(
        feat, lenc, labels, invNf, invNl, pExp, pLS, pCnt);

    finalize_kernel<<<1, 256, 0, stream>>>(pExp, pLS, pCnt, (float*)d_out);
}